// LinearKoopmanLayer_54099408060867
// MI455X (gfx1250) — compile-verified
//
#include <hip/hip_runtime.h>

// LinearKoopmanLayer: per (b,f) scaled 2x2 rotation across T=1024 samples.
// Memory-bound stream: 128 MiB in + 128 MiB out @ 23.3 TB/s ~= 11.5 us floor.
// Data path (CDNA5): global_load_async_to_lds_b128 -> s_wait_asynccnt ->
//                    ds_load_b128 -> v_fma -> non-temporal global_store_b128

typedef __attribute__((ext_vector_type(4))) float v4f;

#define T_LEN   1024                 // samples per row (reference fixes T = 1024)
#define ROW_B   (T_LEN * 4)          // row size in bytes (4096)
#define BLOCK   256                  // 8 wave32; one float4 column pair per thread

__global__ __launch_bounds__(BLOCK) void koopman_rot_kernel(
    const float* __restrict__ x,
    const float* __restrict__ delta_t,
    const float* __restrict__ amplitudes,
    const float* __restrict__ frequencies,
    float* __restrict__ out)
{
    // LDS: row0 (1024 f32) | row1 (1024 f32) | {c, s} broadcast slot
    __shared__ __align__(16) float smem[2 * T_LEN + 4];

    const int f   = blockIdx.x;          // feature   (grid.x = F)
    const int b   = blockIdx.y;          // batch     (grid.y = B)
    const int F   = gridDim.x;
    const int tid = threadIdx.x;

    // rows 2f and 2f+1 of batch b are contiguous: base = (b*F + f) * 2*T
    const size_t rowbase = ((size_t)b * F + f) * (2 * T_LEN);

    const float* g0 = x + rowbase + tid * 4;        // x0 chunk (16B); x1 at +4096B
    unsigned l0 = (unsigned)(unsigned long long)(&smem[tid * 4]);
    unsigned l1 = (unsigned)(unsigned long long)(&smem[T_LEN + tid * 4]);

    // Async global->LDS copies (ASYNCcnt-tracked, no VGPR landing zone).
    // Second row folded into the signed 24-bit instruction offset.
    asm volatile(
        "global_load_async_to_lds_b128 %0, %2, off\n\t"
        "global_load_async_to_lds_b128 %1, %2, off offset:4096"
        :
        : "v"(l0), "v"(l1), "v"(g0)
        : "memory");

    // Overlap the (tiny) per-block transcendental work with copies in flight.
    if (tid == 0) {
        float dt  = delta_t[b];
        float a   = __powf(amplitudes[f], dt);
        float ang = frequencies[f] * dt;
        float sv, cv;
        __sincosf(ang, &sv, &cv);
        smem[2 * T_LEN]     = a * cv;   // c
        smem[2 * T_LEN + 1] = a * sv;   // s
    }
    __syncthreads();                    // publish c,s (tile region is lane-private)

    const float c = smem[2 * T_LEN];
    const float s = smem[2 * T_LEN + 1];

    // Wait for our own async copies; each lane reads exactly the bytes it issued.
    asm volatile("s_wait_asynccnt 0" ::: "memory");

    v4f x0 = *(const v4f*)&smem[tid * 4];
    v4f x1 = *(const v4f*)&smem[T_LEN + tid * 4];

    v4f o0 = c * x0 - s * x1;           // contracts to v_fma
    v4f o1 = s * x0 + c * x1;

    // Streaming outputs (never re-read; 256 MB stream > 192 MB L2) -> NT stores.
    float* q0 = out + rowbase + tid * 4;
    __builtin_nontemporal_store(o0, (v4f*)q0);
    __builtin_nontemporal_store(o1, (v4f*)(q0 + T_LEN));
}

extern "C" void kernel_launch(void* const* d_in, const int* in_sizes, int n_in,
                              void* d_out, int out_size, void* d_ws, size_t ws_size,
                              hipStream_t stream) {
    const float* x    = (const float*)d_in[0];   // (B, 2F, T) f32
    const float* dt   = (const float*)d_in[1];   // (B,)
    const float* amp  = (const float*)d_in[2];   // (F,)
    const float* freq = (const float*)d_in[3];   // (F,)
    float* out = (float*)d_out;

    const int B = in_sizes[1];                   // from delta_t
    const int F = in_sizes[2];                   // from amplitudes

    dim3 grid(F, B);                             // f = blockIdx.x, b = blockIdx.y
    dim3 block(BLOCK);
    koopman_rot_kernel<<<grid, block, 0, stream>>>(x, dt, amp, freq, out);
}